// HoughCriterion_61649960567092
// MI455X (gfx1250) — compile-verified
//
#include <hip/hip_runtime.h>
#include <hip/hip_bf16.h>

// HoughCriterion (CenterNet gaussian focal loss) for MI455X / gfx1250.
// B=8, N=32 boxes, C=80 classes; levels 128x128, 64x64, 32x32.
// Pass A: bandwidth-bound base sweep (target==0 term) with WMMA wave reduction.
// Pass B: sparse per-box-window corrections (deterministic, no atomics).
// Pass C: per-channel loss assembly + scalar tail.

typedef __attribute__((ext_vector_type(2))) float v2f;
typedef __attribute__((ext_vector_type(8))) float v8f;

#define B_   8
#define N_   32
#define C_   80
#define BC_  (B_ * C_)          // 640 channels
#define NBLK_B (3 * B_ * N_)    // 768 correction blocks
#define EPSG 2.2204460492503131e-16f

__device__ __forceinline__ float sigmoid_clip(float h) {
  float p = 1.0f / (1.0f + __expf(-h));
  return fminf(fmaxf(p, 1.0e-4f), 1.0f - 1.0e-4f);
}
// t==0 neg-loss term: log(1-p) * p^2   (neg_w == 1)
__device__ __forceinline__ float neg_base(float h) {
  float p = sigmoid_clip(h);
  return __logf(1.0f - p) * p * p;
}

// Full 32-lane sum. Uses V_WMMA_F32_16X16X4_F32: A holds per-lane partials in
// the K=0 (lanes 0-15) / K=2 (lanes 16-31) slots, B = ones(4x16), C = 0.
// D[m][n] = partial[m] + partial[m+16]; summing this lane's 8 D VGPRs gives
// sum of 8 row-pairs; one shfl_xor(16) completes the reduction (all lanes).
__device__ __forceinline__ float wave_sum32(float v) {
#if __has_builtin(__builtin_amdgcn_wmma_f32_16x16x4_f32)
  v2f a; a[0] = v;    a[1] = 0.0f;
  v2f b; b[0] = 1.0f; b[1] = 1.0f;
  v8f c = {};
  v8f d = __builtin_amdgcn_wmma_f32_16x16x4_f32(
      /*neg_a=*/false, a, /*neg_b=*/false, b,
      /*c_mod=*/(short)0, c, /*reuse_a=*/false, /*reuse_b=*/false);
  float s = ((d[0] + d[1]) + (d[2] + d[3])) + ((d[4] + d[5]) + (d[6] + d[7]));
  s += __shfl_xor(s, 16, 32);
  return s;
#else
  #pragma unroll
  for (int m = 16; m > 0; m >>= 1) v += __shfl_xor(v, m, 32);
  return v;
#endif
}

// ---------------- Pass A: base sweep, one block per (b,c) channel ----------
__global__ __launch_bounds__(1024)
void hough_passA(const float* __restrict__ h0, const float* __restrict__ h1,
                 const float* __restrict__ h2, float* __restrict__ nlA) {
  const int ch  = blockIdx.x;          // b*C + c
  const int tid = threadIdx.x;
  float acc = 0.0f;

  const float4* q0 = (const float4*)(h0 + (size_t)ch * 16384);
  #pragma unroll
  for (int i = 0; i < 4; ++i) {        // 16384 floats = 4 * 1024 * float4
    float4 v = q0[i * 1024 + tid];
    acc += neg_base(v.x) + neg_base(v.y) + neg_base(v.z) + neg_base(v.w);
  }
  const float4* q1 = (const float4*)(h1 + (size_t)ch * 4096);
  {
    float4 v = q1[tid];                // 4096 floats = 1024 * float4
    acc += neg_base(v.x) + neg_base(v.y) + neg_base(v.z) + neg_base(v.w);
  }
  acc += neg_base(h2[(size_t)ch * 1024 + tid]);   // 1024 floats

  float ws = wave_sum32(acc);
  __shared__ float red[32];
  const int wave = tid >> 5, lane = tid & 31;
  if (lane == 0) red[wave] = ws;
  __syncthreads();
  if (wave == 0) {
    float s2 = wave_sum32(red[lane]);  // whole wave 0 active -> EXEC all 1s
    if (lane == 0) nlA[ch] = s2;
  }
}

// ---------------- Pass B: per-(box, level) corrections ---------------------
// corr[blk*4] = channel id (bit-cast), then {dpos, dpl, dnl}.
__global__ __launch_bounds__(256)
void hough_passB(const float* __restrict__ h0, const float* __restrict__ h1,
                 const float* __restrict__ h2, const float* __restrict__ boxes,
                 const int* __restrict__ labels, float* __restrict__ corr) {
  const int blk   = blockIdx.x;             // 0..767
  const int level = blk / (B_ * N_);
  const int bn    = blk % (B_ * N_);
  const int b     = bn / N_;
  const int j     = bn % N_;
  const int S     = (level == 0) ? 128 : (level == 1) ? 64 : 32;
  const float* heat = (level == 0) ? h0 : (level == 1) ? h1 : h2;

  __shared__ int   sx[N_], sy[N_], sr[N_], slab[N_];
  __shared__ float sinv[N_];
  if (threadIdx.x < N_) {
    const int k = threadIdx.x;
    const float* bx = boxes + ((size_t)b * N_ + k) * 4;
    const float fS = (float)S;
    sx[k] = (int)floorf(bx[0] * fS);
    sy[k] = (int)floorf(bx[1] * fS);
    float bw = bx[2] * fS, bh = bx[3] * fS;
    int r = (int)floorf((bw + bh) * 0.25f);
    if (r < 1) r = 1;
    sr[k] = r;
    float sigma = (float)(2 * r + 1) / 6.0f;
    sinv[k] = 1.0f / (2.0f * sigma * sigma);
    slab[k] = labels[b * N_ + k];
  }
  __syncthreads();

  const int c = slab[j], xj = sx[j], yj = sy[j], rj = sr[j];
  const int side = 2 * rj + 1, total = side * side;
  const float* chan = heat + (size_t)(b * C_ + c) * S * S;

  float dpos = 0.0f, dpl = 0.0f, dnl = 0.0f;
  for (int t = threadIdx.x; t < total; t += blockDim.x) {
    const int px = xj - rj + (t % side);
    const int py = yj - rj + (t / side);
    if ((unsigned)px >= (unsigned)S || (unsigned)py >= (unsigned)S) continue;

    // dedup: skip if a lower-index same-class box's window covers this pixel
    bool skip = false;
    for (int k = 0; k < j; ++k)
      if (slab[k] == c && abs(px - sx[k]) <= sr[k] && abs(py - sy[k]) <= sr[k]) { skip = true; break; }
    if (skip) continue;

    // target = max over all same-class windowed gaussians (>= EPS)
    float tmax = 0.0f;
    for (int k = 0; k < N_; ++k) {
      if (slab[k] != c) continue;
      const int dx = px - sx[k], dy = py - sy[k];
      if (abs(dx) > sr[k] || abs(dy) > sr[k]) continue;
      const float g = __expf(-(float)(dx * dx + dy * dy) * sinv[k]);
      if (g >= EPSG) tmax = fmaxf(tmax, g);
    }
    if (tmax <= 0.0f) continue;

    const float h  = chan[py * S + px];
    const float p  = sigmoid_clip(h);
    const float om = 1.0f - p;
    const float base = __logf(om) * p * p;
    if (tmax == 1.0f) {            // positive pixel: swap base term for pos term
      dnl  -= base;
      dpl  += __logf(p) * om * om;
      dpos += 1.0f;
    } else {                       // reweight base by (1-t)^4
      const float u = 1.0f - tmax;
      float w = u * u; w = w * w;
      dnl += base * (w - 1.0f);
    }
  }

  // deterministic block reduction: shuffle within wave, serial over 8 waves
  #pragma unroll
  for (int m = 16; m > 0; m >>= 1) {
    dpos += __shfl_xor(dpos, m, 32);
    dpl  += __shfl_xor(dpl,  m, 32);
    dnl  += __shfl_xor(dnl,  m, 32);
  }
  __shared__ float rp[8], rl[8], rn[8];
  const int wave = threadIdx.x >> 5, lane = threadIdx.x & 31;
  if (lane == 0) { rp[wave] = dpos; rl[wave] = dpl; rn[wave] = dnl; }
  __syncthreads();
  if (threadIdx.x == 0) {
    float sp = 0.0f, sl = 0.0f, sn = 0.0f;
    #pragma unroll
    for (int w = 0; w < 8; ++w) { sp += rp[w]; sl += rl[w]; sn += rn[w]; }
    corr[blk * 4 + 0] = __int_as_float(b * C_ + c);
    corr[blk * 4 + 1] = sp;
    corr[blk * 4 + 2] = sl;
    corr[blk * 4 + 3] = sn;
  }
}

// ---------------- Pass C: assemble per-channel loss, scalar tail -----------
__global__ __launch_bounds__(1024)
void hough_passC(const float* __restrict__ nlA, const float* __restrict__ corr,
                 float* __restrict__ out) {
  __shared__ float sl[1024];
  const int ch = threadIdx.x;
  float loss = 0.0f;
  if (ch < BC_) {
    float nl = nlA[ch], pos = 0.0f, pl = 0.0f;
    for (int i = 0; i < NBLK_B; ++i) {     // fixed order -> deterministic
      if (__float_as_int(corr[i * 4 + 0]) == ch) {
        pos += corr[i * 4 + 1];
        pl  += corr[i * 4 + 2];
        nl  += corr[i * 4 + 3];
      }
    }
    loss = (pos == 0.0f) ? (-nl) : (-(pl + nl) / pos);
    loss = fminf(loss, 10.0f);
  }
  sl[threadIdx.x] = loss;
  __syncthreads();
  #pragma unroll
  for (int s = 512; s > 0; s >>= 1) {
    if (threadIdx.x < s) sl[threadIdx.x] += sl[threadIdx.x + s];
    __syncthreads();
  }
  if (threadIdx.x == 0) {
    const float mean = sl[0] / (float)BC_;
    const float lm = log1pf(mean);
    out[0] = lm / (1.0f + lm);
  }
}

extern "C" void kernel_launch(void* const* d_in, const int* in_sizes, int n_in,
                              void* d_out, int out_size, void* d_ws, size_t ws_size,
                              hipStream_t stream) {
  const float* h0     = (const float*)d_in[0];   // [8,80,128,128]
  const float* h1     = (const float*)d_in[1];   // [8,80,64,64]
  const float* h2     = (const float*)d_in[2];   // [8,80,32,32]
  const float* boxes  = (const float*)d_in[3];   // [8,32,4]
  const int*   labels = (const int*)d_in[4];     // [8,32]
  // d_in[5] image_sizes: cancels out in the math; unused.
  float* out = (float*)d_out;

  float* nlA  = (float*)d_ws;          // 640 floats
  float* corr = nlA + BC_;             // 768*4 floats

  hough_passA<<<BC_,     1024, 0, stream>>>(h0, h1, h2, nlA);
  hough_passB<<<NBLK_B,  256,  0, stream>>>(h0, h1, h2, boxes, labels, corr);
  hough_passC<<<1,       1024, 0, stream>>>(nlA, corr, out);
}